// TruncatedDCT_77266461655518
// MI455X (gfx1250) — compile-verified
//
#include <hip/hip_runtime.h>
#include <math.h>

typedef __attribute__((ext_vector_type(2))) float v2f;
typedef __attribute__((ext_vector_type(8))) float v8f;

#define KCOEF 32
#define NDIM  256
#define LSTR  44   // LDS row stride (floats): 44*2 % 64 = 24, 44*8 % 64 = 32 -> conflict-free

__global__ __launch_bounds__(256) void TruncatedDCT_kernel(
    const float* __restrict__ x, float* __restrict__ out)
{
    __shared__ float Bt[NDIM * LSTR];  // Bt[w][l] = 2*cos(pi*(2w+1)*l/(2N))  (basis, transposed)
    __shared__ float Tl[NDIM * LSTR];  // T[h][l] = (X * B^T)[h][l]

    const int tid  = threadIdx.x;
    const int wave = tid >> 5;
    const int lane = tid & 31;
    const int half = lane >> 4;   // 0: lanes 0-15, 1: lanes 16-31
    const int l16  = lane & 15;

    // ---- build transposed DCT-II basis in LDS (8192 entries / 256 threads) ----
    const float PI = 3.14159265358979323846f;
    for (int i = tid; i < NDIM * KCOEF; i += 256) {
        int w = i >> 5;          // 0..255 (sample index)
        int l = i & 31;          // 0..31  (frequency index)
        float ang = PI * (2.0f * (float)w + 1.0f) * (float)l * (1.0f / (2.0f * (float)NDIM));
        Bt[w * LSTR + l] = 2.0f * cosf(ang);
    }
    __syncthreads();

    const float* X = x + (size_t)blockIdx.x * (size_t)(NDIM * NDIM);

    // ---- Stage 1: T[h][l] = sum_w X[h][w] * Basis[l][w] ----
    // wave handles h-tiles {2*wave, 2*wave+1} (rows h0..h0+31) x n-tiles {0,1}
    const int h0   = wave * 32;
    const int koff = half * 2;                 // K offset within the 16x4 A fragment
    const float* Xr0 = X + (size_t)(h0 + l16) * NDIM + koff;
    const float* Xr1 = Xr0 + 16 * NDIM;

    v8f acc00 = {}; v8f acc01 = {}; v8f acc10 = {}; v8f acc11 = {};

    #pragma unroll 4
    for (int k0 = 0; k0 < NDIM; k0 += 4) {
        // A fragments: A[m][k] = X[h0+m][k0+k]  (v2f per lane, 8B-aligned)
        v2f a0 = *(const v2f*)(Xr0 + k0);
        v2f a1 = *(const v2f*)(Xr1 + k0);
        // B fragments: B[k][n] = Basis[n][k0+k] = Bt[k0+k][n]
        const float* br = &Bt[(k0 + koff) * LSTR];
        v2f b0, b1;
        b0.x = br[l16];            b0.y = br[LSTR + l16];
        b1.x = br[16 + l16];       b1.y = br[LSTR + 16 + l16];

        acc00 = __builtin_amdgcn_wmma_f32_16x16x4_f32(false, a0, false, b0, (short)0, acc00, false, false);
        acc01 = __builtin_amdgcn_wmma_f32_16x16x4_f32(false, a0, false, b1, (short)0, acc01, false, false);
        acc10 = __builtin_amdgcn_wmma_f32_16x16x4_f32(false, a1, false, b0, (short)0, acc10, false, false);
        acc11 = __builtin_amdgcn_wmma_f32_16x16x4_f32(false, a1, false, b1, (short)0, acc11, false, false);
    }

    // spill T tiles to LDS: D-layout row = i + 8*half, col = l16 (+ n-tile*16)
    #pragma unroll
    for (int i = 0; i < 8; ++i) {
        int r = h0 + i + 8 * half;
        Tl[r * LSTR + l16]               = acc00[i];
        Tl[r * LSTR + 16 + l16]          = acc01[i];
        Tl[(r + 16) * LSTR + l16]        = acc10[i];
        Tl[(r + 16) * LSTR + 16 + l16]   = acc11[i];
    }
    __syncthreads();

    // ---- Stage 2: Y[k][l] = sum_h Basis[k][h] * T[h][l]  (32x32 out, 4 tiles) ----
    if (wave < 4) {  // wave-uniform branch: EXEC stays all-ones within active waves
        const int m0 = (wave >> 1) * 16;
        const int n0 = (wave & 1) * 16;
        v8f acc = {};
        #pragma unroll 4
        for (int k0 = 0; k0 < NDIM; k0 += 4) {
            int kk = k0 + koff;
            v2f a, b;
            // A[m][k] = Basis[m0+m][kk] = Bt[kk][m0+m]
            a.x = Bt[kk * LSTR + m0 + l16];
            a.y = Bt[(kk + 1) * LSTR + m0 + l16];
            // B[k][n] = T[kk][n0+n]
            b.x = Tl[kk * LSTR + n0 + l16];
            b.y = Tl[(kk + 1) * LSTR + n0 + l16];
            acc = __builtin_amdgcn_wmma_f32_16x16x4_f32(false, a, false, b, (short)0, acc, false, false);
        }
        float* Y = out + (size_t)blockIdx.x * (KCOEF * KCOEF);
        #pragma unroll
        for (int i = 0; i < 8; ++i)
            Y[(m0 + i + 8 * half) * KCOEF + n0 + l16] = acc[i];
    }
}

extern "C" void kernel_launch(void* const* d_in, const int* in_sizes, int n_in,
                              void* d_out, int out_size, void* d_ws, size_t ws_size,
                              hipStream_t stream) {
    const float* x = (const float*)d_in[0];
    float* out = (float*)d_out;
    int slices = in_sizes[0] / (NDIM * NDIM);   // B*C = 1024
    TruncatedDCT_kernel<<<slices, 256, 0, stream>>>(x, out);
}